// GCNLayer_28647431864950
// MI455X (gfx1250) — compile-verified
//
#include <hip/hip_runtime.h>

// GCNConv forward for MI455X (gfx1250, wave32).
//   deg -> dinv -> xw = x@W (WMMA f32 16x16x4) -> out = b + dinv^2*xw -> edge scatter atomics
//
// Sizes (fixed by the reference):
#define N_NODES 100000
#define N_EDGES 1600000
#define C_IN 64
#define C_OUT 64

typedef __attribute__((ext_vector_type(2))) float v2f;
typedef __attribute__((ext_vector_type(8))) float v8f;

// ---------------------------------------------------------------- kernel 0: zero degree
__global__ __launch_bounds__(256) void gcn_zero_deg(int* __restrict__ deg) {
    int i = blockIdx.x * 256 + threadIdx.x;
    if (i < N_NODES) deg[i] = 0;
}

// ---------------------------------------------------------------- kernel 1: degree histogram
__global__ __launch_bounds__(256) void gcn_degree(const int* __restrict__ col,
                                                  int* __restrict__ deg) {
    int e = blockIdx.x * 256 + threadIdx.x;
    if (e < N_EDGES) atomicAdd(&deg[col[e]], 1);
}

// ---------------------------------------------------------------- kernel 2: dinv = rsqrt(deg+1)
__global__ __launch_bounds__(256) void gcn_dinv(const int* __restrict__ deg,
                                                float* __restrict__ dinv) {
    int i = blockIdx.x * 256 + threadIdx.x;
    if (i < N_NODES) {
        // +1 for the self loop; deg_hat >= 1 always, so no zero guard needed.
        dinv[i] = rsqrtf((float)deg[i] + 1.0f);
    }
}

// ---------------------------------------------------------------- kernel 3: xw = x @ W  (WMMA f32)
// Block = 256 threads = 8 waves. Each wave computes one 16x16 tile of xw.
// Wave w: row tile = (w>>2), col tile = (w&3)  ->  block covers 32 rows x 64 cols.
// Grid = N_NODES/32 = 3125 blocks (exact).
__global__ __launch_bounds__(256) void gcn_xw_wmma(const float* __restrict__ x,
                                                   const float* __restrict__ W,
                                                   float* __restrict__ xw) {
    __shared__ float sW[C_IN * C_OUT];  // 16 KB, reused by all 8 waves

    const int tid = threadIdx.x;
    // Cooperative 16 KB load of W into LDS: 1024 float4 / 256 threads = 4 each.
    {
        const float4* Wv = (const float4*)W;
        float4* sWv = (float4*)sW;
#pragma unroll
        for (int i = 0; i < 4; ++i) sWv[tid + 256 * i] = Wv[tid + 256 * i];
    }
    __syncthreads();

    const int wave = tid >> 5;
    const int lane = tid & 31;
    const int m0 = blockIdx.x * 32 + (wave >> 2) * 16;  // output row tile base
    const int n0 = (wave & 3) * 16;                     // output col tile base
    const int mh = lane & 15;                           // M (for A) / N (for B,C,D)
    const int kb = (lane >> 4) << 1;                    // lane-half K offset {0,2}

    // A (16x4 f32): lane holds x[m0+mh][k+kb .. k+kb+1]
    const float* arow = x + (size_t)(m0 + mh) * C_IN + kb;
    // B (4x16 f32): lane holds W[k+kb .. k+kb+1][n0+mh]
    const float* bcol = sW + (size_t)kb * C_OUT + n0 + mh;

    v8f acc = {};
#pragma unroll
    for (int k = 0; k < C_IN; k += 4) {
        v2f a, b;
        a.x = arow[k];
        a.y = arow[k + 1];
        b.x = bcol[k * C_OUT];
        b.y = bcol[k * C_OUT + C_OUT];
        // D = A x B + C  (V_WMMA_F32_16X16X4_F32)
        acc = __builtin_amdgcn_wmma_f32_16x16x4_f32(
            /*neg_a=*/false, a, /*neg_b=*/false, b,
            /*c_mod=*/(short)0, acc, /*reuse_a=*/false, /*reuse_b=*/false);
    }

    // C/D layout: VGPR r -> M = m0 + r + 8*(lane>>4), N = n0 + (lane&15)
    float* drow = xw + (size_t)(m0 + 8 * (lane >> 4)) * C_OUT + n0 + mh;
#pragma unroll
    for (int r = 0; r < 8; ++r) drow[r * C_OUT] = acc[r];
}

// ---------------------------------------------------------------- kernel 4: out = b + dinv^2 * xw
// One thread per float2 (channel pair): N_NODES*32 threads.
__global__ __launch_bounds__(256) void gcn_selfloop_bias(const float* __restrict__ xw,
                                                         const float* __restrict__ dinv,
                                                         const float* __restrict__ b,
                                                         float* __restrict__ out) {
    int idx = blockIdx.x * 256 + threadIdx.x;  // float2 index
    if (idx < N_NODES * (C_OUT / 2)) {
        int n = idx >> 5;       // node
        int c2 = idx & 31;      // channel pair
        float di = dinv[n];
        float s = di * di;      // self-loop norm = dinv[i] * 1 * dinv[i]
        float2 xv = ((const float2*)xw)[idx];
        float2 bv = ((const float2*)b)[c2];
        float2 o;
        o.x = bv.x + s * xv.x;
        o.y = bv.y + s * xv.y;
        ((float2*)out)[idx] = o;
    }
}

// ---------------------------------------------------------------- kernel 5: edge scatter-add
// One wave32 per edge; lane l handles channels {2l, 2l+1}. Gather of xw[row] is one
// fully coalesced 256B wave access; scatter uses hardware global_atomic_add_f32.
// xw and out both fit in the 192MB L2, so this is L2-resident traffic.
__global__ __launch_bounds__(256) void gcn_edge_scatter(const int* __restrict__ row,
                                                        const int* __restrict__ col,
                                                        const float* __restrict__ xw,
                                                        const float* __restrict__ dinv,
                                                        float* __restrict__ out) {
    const int e = blockIdx.x * 8 + (threadIdx.x >> 5);  // 8 edges per 256-thread block
    if (e >= N_EDGES) return;                           // uniform per wave
    const int lane = threadIdx.x & 31;
    const int r = row[e];
    const int c = col[e];
    const float norm = dinv[r] * dinv[c];
    const float2 v = ((const float2*)(xw + (size_t)r * C_OUT))[lane];
    float* dst = out + (size_t)c * C_OUT + lane * 2;
    unsafeAtomicAdd(dst, v.x * norm);
    unsafeAtomicAdd(dst + 1, v.y * norm);
}

// ---------------------------------------------------------------- launch
extern "C" void kernel_launch(void* const* d_in, const int* in_sizes, int n_in,
                              void* d_out, int out_size, void* d_ws, size_t ws_size,
                              hipStream_t stream) {
    const float* x = (const float*)d_in[0];       // [N_NODES, C_IN]
    const int* ei = (const int*)d_in[1];          // [2, N_EDGES]
    const float* W = (const float*)d_in[2];       // [C_IN, C_OUT]
    const float* b = (const float*)d_in[3];       // [C_OUT]
    float* out = (float*)d_out;                   // [N_NODES, C_OUT]

    const int* row = ei;             // sources
    const int* col = ei + N_EDGES;   // targets

    // Workspace layout (all offsets 16B-aligned):
    //   deg  : int  [N_NODES]            @ 0         (400,000 B)
    //   dinv : float[N_NODES]            @ 400,000   (400,000 B)
    //   xw   : float[N_NODES*C_OUT]      @ 800,000   (25.6 MB)
    char* ws = (char*)d_ws;
    int* deg = (int*)(ws);
    float* dinv = (float*)(ws + (size_t)N_NODES * 4);
    float* xw = (float*)(ws + (size_t)N_NODES * 8);

    gcn_zero_deg<<<(N_NODES + 255) / 256, 256, 0, stream>>>(deg);
    gcn_degree<<<(N_EDGES + 255) / 256, 256, 0, stream>>>(col, deg);
    gcn_dinv<<<(N_NODES + 255) / 256, 256, 0, stream>>>(deg, dinv);
    gcn_xw_wmma<<<N_NODES / 32, 256, 0, stream>>>(x, W, xw);
    gcn_selfloop_bias<<<(N_NODES * (C_OUT / 2) + 255) / 256, 256, 0, stream>>>(xw, dinv, b, out);
    gcn_edge_scatter<<<N_EDGES / 8, 256, 0, stream>>>(row, col, xw, dinv, out);
}